// BosonSamplerTorch_12163347383027
// MI455X (gfx1250) — compile-verified
//
#include <hip/hip_runtime.h>

// Boson-sampling Ryser permanent on gfx1250 using V_WMMA_F32_16X16X4_F32.
// perm(A) = sum_{S} (-1)^{n-|S|} prod_i (sum_{j in S} A[i,j]),  n = 18, B = 8.
// Row sums over subsets are a real fp32 matmul (mask is 0/1), done on the
// matrix pipe; per-subset row products + sign accumulation on VALU.

#define NN 18
#define NSUB (1u << 18)
#define BLOCKS_X 64
#define WAVES_PER_BLOCK 8
#define THREADS (WAVES_PER_BLOCK * 32)
#define CHUNKS_PER_BATCH (NSUB / 16)                               // 16384
#define CHUNKS_PER_WAVE (CHUNKS_PER_BATCH / (BLOCKS_X * WAVES_PER_BLOCK)) // 32

typedef float v2f __attribute__((ext_vector_type(2)));
typedef float v8f __attribute__((ext_vector_type(8)));

__global__ __launch_bounds__(THREADS)
void ryser_wmma_kernel(const float* __restrict__ Ar,
                       const float* __restrict__ Ai,
                       float* __restrict__ partials) {
  const int b    = blockIdx.y;          // batch 0..7
  const int lane = threadIdx.x & 31;
  const int wave = threadIdx.x >> 5;
  const int half = lane >> 4;           // 0: K even pair low, 1: K high pair
  const int t    = lane & 15;           // row (A-frag) / subset-in-chunk (B-frag, C/D N)

  const float* ar = Ar + b * NN * NN;
  const float* ai = Ai + b * NN * NN;

  // Rows 16,17 (ragged beyond the 16x16 WMMA M-tile) kept in LDS.
  __shared__ float a16r[NN], a16i[NN], a17r[NN], a17i[NN];
  __shared__ float wred[WAVES_PER_BLOCK][2];
  if (threadIdx.x < NN) {
    a16r[threadIdx.x] = ar[16 * NN + threadIdx.x];
    a16i[threadIdx.x] = ai[16 * NN + threadIdx.x];
    a17r[threadIdx.x] = ar[17 * NN + threadIdx.x];
    a17i[threadIdx.x] = ai[17 * NN + threadIdx.x];
  }
  __syncthreads();

  // A-operand fragments (16x4 fp32 per K-chunk), rows 0..15, cols padded to 20.
  // Layout: lane l -> M = l&15 ; VGPR v -> K = 2*(l>>4) + v.
  v2f afr[5], afi[5];
#pragma unroll
  for (int kc = 0; kc < 5; ++kc) {
    const int j0 = 4 * kc + 2 * half;
    v2f r, im;
    r.x  = (j0     < NN) ? ar[t * NN + j0]     : 0.0f;
    r.y  = (j0 + 1 < NN) ? ar[t * NN + j0 + 1] : 0.0f;
    im.x = (j0     < NN) ? ai[t * NN + j0]     : 0.0f;
    im.y = (j0 + 1 < NN) ? ai[t * NN + j0 + 1] : 0.0f;
    afr[kc] = r;
    afi[kc] = im;
  }

  float accr = 0.0f, acci = 0.0f;
  const unsigned waveGlobal = (unsigned)blockIdx.x * WAVES_PER_BLOCK + (unsigned)wave;

  for (int c = 0; c < CHUNKS_PER_WAVE; ++c) {
    const unsigned s0 = (waveGlobal * CHUNKS_PER_WAVE + (unsigned)c) * 16u;
    const unsigned s  = s0 + (unsigned)t;   // this lane's subset (as B-frag N / C-frag N)

    // Row sums rs[M, s] for M=0..15 via chained fp32 WMMAs (real & imag).
    v8f cr = {};
    v8f ci = {};
#pragma unroll
    for (int kc = 0; kc < 5; ++kc) {
      const int j0 = 4 * kc + 2 * half;   // this lane's K index (B-frag layout)
      v2f bm;
      bm.x = (j0     < NN) ? (float)((s >> j0) & 1u)       : 0.0f;
      bm.y = (j0 + 1 < NN) ? (float)((s >> (j0 + 1)) & 1u) : 0.0f;
      cr = __builtin_amdgcn_wmma_f32_16x16x4_f32(false, afr[kc], false, bm,
                                                 (short)0, cr, false, false);
      ci = __builtin_amdgcn_wmma_f32_16x16x4_f32(false, afi[kc], false, bm,
                                                 (short)0, ci, false, false);
    }

    // Complex product of this lane's 8 row sums (M = 8*half + r, N = t).
    float pr = 1.0f, pi = 0.0f;
#pragma unroll
    for (int r = 0; r < 8; ++r) {
      const float xr = cr[r], xi = ci[r];
      const float nr = pr * xr - pi * xi;
      const float ni = pr * xi + pi * xr;
      pr = nr; pi = ni;
    }
    // Partner lane (lane ^ 16) holds the complementary 8 rows for the same subset.
    const float qr = __shfl_xor(pr, 16, 32);
    const float qi = __shfl_xor(pi, 16, 32);
    float fr = pr * qr - pi * qi;
    float fi = pr * qi + pi * qr;

    // Rows 16, 17 via VALU FMAs (LDS broadcast reads).
    float r16r = 0.0f, r16i = 0.0f, r17r = 0.0f, r17i = 0.0f;
#pragma unroll
    for (int j = 0; j < NN; ++j) {
      const float bit = (float)((s >> j) & 1u);
      r16r = fmaf(bit, a16r[j], r16r);
      r16i = fmaf(bit, a16i[j], r16i);
      r17r = fmaf(bit, a17r[j], r17r);
      r17i = fmaf(bit, a17i[j], r17i);
    }
    float tr = fr * r16r - fi * r16i;
    float ti = fr * r16i + fi * r16r;
    fr = tr * r17r - ti * r17i;
    fi = tr * r17i + ti * r17r;

    // n = 18 even -> sign = (-1)^|S|.
    const float sign = (__popc(s) & 1) ? -1.0f : 1.0f;
    accr = fmaf(sign, fr, accr);
    acci = fmaf(sign, fi, acci);
  }

  // Wave reduction. Each subset was accumulated by 2 lanes (lane and lane^16);
  // the duplicate factor of 2 is removed in the finalize kernel.
#pragma unroll
  for (int off = 16; off >= 1; off >>= 1) {
    accr += __shfl_xor(accr, off, 32);
    acci += __shfl_xor(acci, off, 32);
  }
  if (lane == 0) { wred[wave][0] = accr; wred[wave][1] = acci; }
  __syncthreads();
  if (threadIdx.x == 0) {
    float sr = 0.0f, si = 0.0f;
    for (int w = 0; w < WAVES_PER_BLOCK; ++w) { sr += wred[w][0]; si += wred[w][1]; }
    partials[((unsigned)b * BLOCKS_X + blockIdx.x) * 2 + 0] = sr;
    partials[((unsigned)b * BLOCKS_X + blockIdx.x) * 2 + 1] = si;
  }
}

__global__ __launch_bounds__(512)
void finalize_kernel(const float* __restrict__ partials,
                     const float* __restrict__ Ar,
                     const float* __restrict__ Ai,
                     float* __restrict__ out) {
  const int tid = threadIdx.x;     // 512 threads = 8 batches x 64 partials
  const int b   = tid >> 6;
  const int i   = tid & 63;

  float sr = partials[(b * BLOCKS_X + i) * 2 + 0];
  float si = partials[(b * BLOCKS_X + i) * 2 + 1];
#pragma unroll
  for (int off = 16; off >= 1; off >>= 1) {
    sr += __shfl_xor(sr, off, 32);
    si += __shfl_xor(si, off, 32);
  }
  __shared__ float red[16][2];
  const int wv = tid >> 5;
  if ((tid & 31) == 0) { red[wv][0] = sr; red[wv][1] = si; }
  __syncthreads();

  if (i == 0) {
    const float pr  = (red[2 * b][0] + red[2 * b + 1][0]) * 0.5f; // undo lane-half dup
    const float pim = (red[2 * b][1] + red[2 * b + 1][1]) * 0.5f;
    const float abs2 = pr * pr + pim * pim;

    // classical = prod |A|^2 (underflows fp32 to 0, same as reference fp32 prod)
    float cl = 1.0f;
    const float* ar = Ar + b * NN * NN;
    const float* ai = Ai + b * NN * NN;
    for (int k = 0; k < NN * NN; ++k) cl *= ar[k] * ar[k] + ai[k] * ai[k];

    const float MU = (float)(0.85 * 0.98 * 0.98 * 0.99);   // 0.8081766
    out[b] = MU * abs2 + (1.0f - MU) * cl + 1.8e-5f;        // + dark counts (1e-6 * 18)
  }
}

extern "C" void kernel_launch(void* const* d_in, const int* in_sizes, int n_in,
                              void* d_out, int out_size, void* d_ws, size_t ws_size,
                              hipStream_t stream) {
  (void)in_sizes; (void)n_in; (void)out_size; (void)ws_size;
  const float* Ar = (const float*)d_in[0];
  const float* Ai = (const float*)d_in[1];
  float* out      = (float*)d_out;
  float* partials = (float*)d_ws;   // 8 * 64 * 2 floats = 4 KB

  dim3 grid(BLOCKS_X, 8);
  ryser_wmma_kernel<<<grid, THREADS, 0, stream>>>(Ar, Ai, partials);
  finalize_kernel<<<1, 512, 0, stream>>>(partials, Ar, Ai, out);
}